// CausalWaveletFieldAttentionByte_82729660056316
// MI455X (gfx1250) — compile-verified
//
#include <hip/hip_runtime.h>
#include <hip/hip_bf16.h>
#include <math.h>
#include <cstdint>

// Problem constants
#define BATCH   8
#define NSEQ    2048
#define DMODEL  1024
#define NHEADS  16
#define HEADDIM 64
#define NSCALES 11
#define MROWS   (BATCH * NSEQ)        // 16384

typedef float v2f  __attribute__((ext_vector_type(2)));
typedef float v8f  __attribute__((ext_vector_type(8)));
typedef unsigned int u32x4 __attribute__((ext_vector_type(4)));
typedef int   i32x8 __attribute__((ext_vector_type(8)));
typedef int   i32x4 __attribute__((ext_vector_type(4)));

// ---------------------------------------------------------------------------
// Kernel 0: softmax prep for scale gains [11,16] (over scales) and head
// coupling [16,16] (over last dim)
// ---------------------------------------------------------------------------
__global__ void prep_kernel(const float* __restrict__ sg,
                            const float* __restrict__ fc,
                            float* __restrict__ gains,
                            float* __restrict__ coup) {
    int t = threadIdx.x;
    if (t < 16) {
        float m = -1e30f;
        for (int s = 0; s < NSCALES; ++s) m = fmaxf(m, sg[s * 16 + t]);
        float e[NSCALES]; float sum = 0.f;
        for (int s = 0; s < NSCALES; ++s) { e[s] = __expf(sg[s * 16 + t] - m); sum += e[s]; }
        float inv = 1.0f / sum;
        for (int s = 0; s < NSCALES; ++s) gains[s * 16 + t] = e[s] * inv;

        float m2 = -1e30f;
        for (int j = 0; j < 16; ++j) m2 = fmaxf(m2, fc[t * 16 + j]);
        float e2[16]; float s2 = 0.f;
        for (int j = 0; j < 16; ++j) { e2[j] = __expf(fc[t * 16 + j] - m2); s2 += e2[j]; }
        float inv2 = 1.0f / s2;
        for (int j = 0; j < 16; ++j) coup[t * 16 + j] = e2[j] * inv2;
    }
}

// ---------------------------------------------------------------------------
// GEMM1 (WMMA f32 16x16x4): [16384 x 1024] x [1024 x 3072]^T
//   cols [0,1024)    -> k    (qkv_w rows 1024+c)
//   cols [1024,2048) -> v    (qkv_w rows 1024+c, since 2048+(c-1024)=1024+c)
//   cols [2048,3072) -> sigmoid(gate) (gate_w rows c-2048 + gate_b)
// Block tile 64(M) x 32(cols), K panels of 32; 8 waves, one 16x16 C tile each.
// LDS pitch 36 floats: conflict-free frag reads + 16B-aligned b128 staging.
// ---------------------------------------------------------------------------
__global__ __launch_bounds__(256) void gemm1_kernel(
    const float* __restrict__ x,
    const float* __restrict__ qkv_w,
    const float* __restrict__ qkv_b,
    const float* __restrict__ gate_w,
    const float* __restrict__ gate_b,
    float* __restrict__ kbuf,
    float* __restrict__ vbuf,
    float* __restrict__ gatebuf)
{
    __shared__ alignas(16) float sA[64][36];
    __shared__ alignas(16) float sB[32][36];
    const int tid  = threadIdx.x;
    const int m0   = blockIdx.x * 64;
    const int c0   = blockIdx.y * 32;
    const int wv   = tid >> 5;
    const int lane = tid & 31;
    const int half = lane >> 4;
    const int r    = lane & 15;
    const int mw   = (wv & 3) * 16;
    const int cw   = (wv >> 2) * 16;

    // B row pointer for this thread's staging column
    const int bkq = (tid & 7) * 4;
    const int bc  = tid >> 3;
    const int bcg = c0 + bc;
    const float* wrow = (bcg < 2048) ? (qkv_w + (size_t)(1024 + bcg) * DMODEL)
                                     : (gate_w + (size_t)(bcg - 2048) * DMODEL);

    v8f acc = {};

    for (int k0 = 0; k0 < DMODEL; k0 += 32) {
        // A panel: 64 x 32, float4 per thread (coalesced b128)
        #pragma unroll
        for (int i = 0; i < 2; ++i) {
            int lin = tid + i * 256;           // 0..511
            int kq  = (lin & 7) * 4;
            int m   = lin >> 3;
            float4 vx = *(const float4*)(x + (size_t)(m0 + m) * DMODEL + k0 + kq);
            *(float4*)&sA[m][kq] = vx;
        }
        // B panel: 32 cols x 32 k, one float4 per thread
        *(float4*)&sB[bc][bkq] = *(const float4*)(wrow + k0 + bkq);
        __syncthreads();

        #pragma unroll
        for (int k4 = 0; k4 < 32; k4 += 4) {
            // A 16x4: lanes 0-15 hold K=k4+0/1, lanes 16-31 hold K=k4+2/3
            v2f a, b;
            a.x = sA[mw + r][k4 + 2 * half];
            a.y = sA[mw + r][k4 + 2 * half + 1];
            b.x = sB[cw + r][k4 + 2 * half];
            b.y = sB[cw + r][k4 + 2 * half + 1];
            acc = __builtin_amdgcn_wmma_f32_16x16x4_f32(
                false, a, false, b, (short)0, acc, false, false);
        }
        __syncthreads();
    }

    #pragma unroll
    for (int rr = 0; rr < 8; ++rr) {
        int row = mw + rr + 8 * half;
        int col = cw + r;
        int m   = m0 + row;
        int cg  = c0 + col;
        float val = acc[rr];
        if (cg < 1024) {
            kbuf[(size_t)m * DMODEL + cg] = val + qkv_b[1024 + cg];
        } else if (cg < 2048) {
            int cc = cg - 1024;
            vbuf[(size_t)m * DMODEL + cc] = val + qkv_b[2048 + cc];
        } else {
            int cc = cg - 2048;
            float g = val + gate_b[cc];
            gatebuf[(size_t)m * DMODEL + cc] = 1.0f / (1.0f + __expf(-g));
        }
    }
}

// ---------------------------------------------------------------------------
// k magnitude: kmag[b,n,h] = ||k[b,n,h,:]||_2
// ---------------------------------------------------------------------------
__global__ __launch_bounds__(256) void kmag_kernel(const float* __restrict__ kbuf,
                                                   float* __restrict__ kmag) {
    int idx = blockIdx.x * 256 + threadIdx.x;   // B*N*H
    int h  = idx & 15;
    int bn = idx >> 4;
    const float4* p = (const float4*)(kbuf + (size_t)bn * DMODEL + h * HEADDIM);
    float s = 0.f;
    #pragma unroll
    for (int i = 0; i < 16; ++i) {
        float4 q = p[i];
        s += q.x * q.x + q.y * q.y + q.z * q.z + q.w * q.w;
    }
    kmag[idx] = sqrtf(s);
}

// ---------------------------------------------------------------------------
// Causal multiscale D4 conv over the sequence, per (b, channel) row.
// ---------------------------------------------------------------------------
__global__ __launch_bounds__(256) void conv_kernel(
    const float* __restrict__ vbuf, const float* __restrict__ kmag,
    const float* __restrict__ gains, float* __restrict__ convout /* [B,C,N] */) {
    __shared__ float row[NSEQ];
    int bid = blockIdx.x;            // B*C
    int b = bid >> 10;
    int c = bid & 1023;
    int h = c >> 6;

    #pragma unroll
    for (int i = 0; i < 8; ++i) {
        int n = threadIdx.x + i * 256;
        size_t bn = (size_t)b * NSEQ + n;
        row[n] = vbuf[bn * DMODEL + c] * kmag[bn * NHEADS + h];
    }
    __syncthreads();

    float g[NSCALES];
    #pragma unroll
    for (int j = 0; j < NSCALES; ++j) g[j] = gains[j * 16 + h];

    const float d40 =  0.4829629131445341f;
    const float d41 =  0.8365163037378079f;
    const float d42 =  0.2241438680420134f;
    const float d43 = -0.1294095225512604f;

    #pragma unroll
    for (int i = 0; i < 8; ++i) {
        int n = threadIdx.x + i * 256;
        float acc = 0.f;
        #pragma unroll
        for (int j = 0; j < NSCALES; ++j) {
            int d = 1 << j;
            float y = d43 * row[n];
            int n1 = n - d;      if (n1 >= 0) y += d42 * row[n1];
            int n2 = n - 2 * d;  if (n2 >= 0) y += d41 * row[n2];
            int n3 = n - 3 * d;  if (n3 >= 0) y += d40 * row[n3];
            acc += g[j] * y;
        }
        convout[((size_t)b * DMODEL + c) * NSEQ + n] = acc;
    }
}

// ---------------------------------------------------------------------------
// Head coupling + gate, output transposed yT[B, C, N].
// LDS stage: 16 rows x 512 f32 tile from convout — done with the Tensor Data
// Mover when available (D# per CDNA5 ISA), else a plain
// coalesced load loop.
// ---------------------------------------------------------------------------
__global__ __launch_bounds__(256) void couple_gate_kernel(
    const float* __restrict__ convout, const float* __restrict__ gatebuf,
    const float* __restrict__ coup, float* __restrict__ yT) {
    __shared__ alignas(16) float tile[16][512];
    __shared__ float cp[256];
    int bid   = blockIdx.x;          // B * HEADDIM * 4
    int chunk = bid & 3;
    int dd    = (bid >> 2) & 63;
    int b     = bid >> 8;
    int n0    = chunk * 512;

    cp[threadIdx.x] = coup[threadIdx.x];

    const size_t src_off = ((size_t)b * DMODEL + dd) * NSEQ + n0; // row j=0 start

#if __has_builtin(__builtin_amdgcn_tensor_load_to_lds) && __has_builtin(__builtin_amdgcn_s_wait_tensorcnt)
    if (threadIdx.x < 32) {
        // TDM descriptor: 2D tile, dim0 = 512 f32 (contiguous), dim1 = 16 rows,
        // row stride = HEADDIM*NSEQ elements. data_size = 4 bytes.
        unsigned lds_off = (unsigned)(uintptr_t)(&tile[0][0]);      // generic LDS addr truncates to LDS offset
        unsigned long long ga = (unsigned long long)(uintptr_t)(convout + src_off);
        u32x4 g0;
        g0[0] = 1u;                                                  // count=1, user mode, no gather
        g0[1] = lds_off;                                             // D#.lds_addr
        g0[2] = (unsigned)(ga & 0xffffffffu);                        // global_addr[31:0]
        g0[3] = (unsigned)((ga >> 32) & 0x01ffffffu) | (2u << 30);   // global_addr[56:32] | type=2
        i32x8 g1;
        g1[0] = (2 << 16);                   // workgroup_mask=0, data_size=2 (4B), no pad/iterate
        g1[1] = (int)(2048u << 16);          // tensor_dim0[15:0]=2048 (bits 63:48)
        g1[2] = (int)(1024u << 16);          // tensor_dim0[31:16]=0, tensor_dim1[15:0]=1024
        g1[3] = (int)(512u << 16);           // tensor_dim1[31:16]=0, tile_dim0=512
        g1[4] = 16;                          // tile_dim1=16, tile_dim2=0
        g1[5] = (int)(HEADDIM * NSEQ);       // tensor_dim0_stride[31:0] = 131072
        g1[6] = 0;                           // stride bits [47:32] = 0, dim1_stride lo = 0
        g1[7] = 0;
        i32x4 gz  = {0, 0, 0, 0};
        i32x8 gz8 = {0, 0, 0, 0, 0, 0, 0, 0};
        __builtin_amdgcn_tensor_load_to_lds(g0, g1, gz, gz, gz8, 0);
        __builtin_amdgcn_s_wait_tensorcnt(0);
    }
#else
    #pragma unroll
    for (int i = 0; i < 32; ++i) {
        int lin = threadIdx.x + i * 256;   // 16*512
        int j  = lin >> 9;
        int nl = lin & 511;
        tile[j][nl] = convout[src_off + (size_t)j * HEADDIM * NSEQ + nl];
    }
#endif
    __syncthreads();

    #pragma unroll
    for (int t = 0; t < 2; ++t) {
        int nl = threadIdx.x + t * 256;
        int n  = n0 + nl;
        #pragma unroll
        for (int ih = 0; ih < 16; ++ih) {
            float acc = 0.f;
            #pragma unroll
            for (int j = 0; j < 16; ++j) acc += cp[ih * 16 + j] * tile[j][nl];
            int cch = ih * HEADDIM + dd;
            float gte = gatebuf[((size_t)b * NSEQ + n) * DMODEL + cch];
            yT[((size_t)b * DMODEL + cch) * NSEQ + n] = acc * gte;
        }
    }
}

// ---------------------------------------------------------------------------
// GEMM2 (WMMA f32): out = y @ out_w^T + out_b, A sourced from yT[B,C,N]
// (transpose resolved during the LDS stage).
// ---------------------------------------------------------------------------
__global__ __launch_bounds__(256) void gemm2_kernel(
    const float* __restrict__ yT, const float* __restrict__ out_w,
    const float* __restrict__ out_b, float* __restrict__ out) {
    __shared__ alignas(16) float sA[64][36];
    __shared__ alignas(16) float sB[32][36];
    const int tid  = threadIdx.x;
    const int m0   = blockIdx.x * 64;
    const int c0   = blockIdx.y * 32;
    const int b    = m0 >> 11;        // tile never crosses batch (2048 % 64 == 0)
    const int nb   = m0 & 2047;
    const int wv   = tid >> 5;
    const int lane = tid & 31;
    const int half = lane >> 4;
    const int r    = lane & 15;
    const int mw   = (wv & 3) * 16;
    const int cw   = (wv >> 2) * 16;

    const int bkq = (tid & 7) * 4;
    const int bc  = tid >> 3;

    v8f acc = {};

    for (int k0 = 0; k0 < DMODEL; k0 += 32) {
        // A tile: element (m,k) = yT[b, k0+k, nb+m]; float4 along m (coalesced)
        #pragma unroll
        for (int i = 0; i < 2; ++i) {
            int lin = tid + i * 256;          // 0..511
            int mq  = (lin & 15) * 4;
            int k   = lin >> 4;
            float4 vy = *(const float4*)(yT + ((size_t)b * DMODEL + k0 + k) * NSEQ + nb + mq);
            sA[mq + 0][k] = vy.x;
            sA[mq + 1][k] = vy.y;
            sA[mq + 2][k] = vy.z;
            sA[mq + 3][k] = vy.w;
        }
        *(float4*)&sB[bc][bkq] = *(const float4*)(out_w + (size_t)(c0 + bc) * DMODEL + k0 + bkq);
        __syncthreads();

        #pragma unroll
        for (int k4 = 0; k4 < 32; k4 += 4) {
            v2f a, bb;
            a.x  = sA[mw + r][k4 + 2 * half];
            a.y  = sA[mw + r][k4 + 2 * half + 1];
            bb.x = sB[cw + r][k4 + 2 * half];
            bb.y = sB[cw + r][k4 + 2 * half + 1];
            acc = __builtin_amdgcn_wmma_f32_16x16x4_f32(
                false, a, false, bb, (short)0, acc, false, false);
        }
        __syncthreads();
    }

    #pragma unroll
    for (int rr = 0; rr < 8; ++rr) {
        int row = mw + rr + 8 * half;
        int cg  = c0 + cw + r;
        out[(size_t)(m0 + row) * DMODEL + cg] = acc[rr] + out_b[cg];
    }
}

// ---------------------------------------------------------------------------
// Launch
// ---------------------------------------------------------------------------
extern "C" void kernel_launch(void* const* d_in, const int* in_sizes, int n_in,
                              void* d_out, int out_size, void* d_ws, size_t ws_size,
                              hipStream_t stream) {
    const float* x      = (const float*)d_in[0];
    const float* qkv_w  = (const float*)d_in[1];
    const float* qkv_b  = (const float*)d_in[2];
    const float* out_w  = (const float*)d_in[3];
    const float* out_b  = (const float*)d_in[4];
    const float* gate_w = (const float*)d_in[5];
    const float* gate_b = (const float*)d_in[6];
    const float* sg     = (const float*)d_in[7];
    const float* fc     = (const float*)d_in[8];
    float* out = (float*)d_out;

    const size_t MD = (size_t)MROWS * DMODEL;   // 16,777,216 floats (64 MB)
    float* ws      = (float*)d_ws;
    float* gains   = ws;                        // 176 floats
    float* coup    = ws + 256;                  // 256 floats
    float* kbuf    = ws + 1024;                 // [M, D]
    float* vbuf    = kbuf + MD;                 // [M, D]
    float* gatebuf = vbuf + MD;                 // [M, D]
    float* kmag    = gatebuf + MD;              // [B, N, H]
    // Aliased reuse (producers fully consumed before overwrite):
    float* convout = kbuf;                      // conv output [B, C, N]
    float* yT      = vbuf;                      // gated/coupled output [B, C, N]

    prep_kernel<<<1, 32, 0, stream>>>(sg, fc, gains, coup);

    dim3 g1(MROWS / 64, 3072 / 32);             // 256 x 96
    gemm1_kernel<<<g1, 256, 0, stream>>>(x, qkv_w, qkv_b, gate_w, gate_b,
                                         kbuf, vbuf, gatebuf);

    kmag_kernel<<<(BATCH * NSEQ * NHEADS) / 256, 256, 0, stream>>>(kbuf, kmag);

    conv_kernel<<<BATCH * DMODEL, 256, 0, stream>>>(vbuf, kmag, gains, convout);

    couple_gate_kernel<<<BATCH * HEADDIM * 4, 256, 0, stream>>>(convout, gatebuf,
                                                                coup, yT);

    dim3 g2(MROWS / 64, DMODEL / 32);           // 256 x 32
    gemm2_kernel<<<g2, 256, 0, stream>>>(yT, out_w, out_b, out);
}